// SGConv_29978871726569
// MI455X (gfx1250) — compile-verified
//
#include <hip/hip_runtime.h>
#include <hip/hip_bf16.h>

// ---------------------------------------------------------------------------
// SGConv on MI455X (gfx1250):
//   4x SpMM hops (L2-resident gather + f32 atomic scatter) followed by a
//   fp32 WMMA GEMM (V_WMMA_F32_16X16X4_F32) for the final x @ W + b.
// ---------------------------------------------------------------------------

#define N_NODES 100000
#define D_IN    32
#define D_OUT   64

typedef __attribute__((ext_vector_type(2))) float v2f;
typedef __attribute__((ext_vector_type(8))) float v8f;

// ----------------------------- zero-fill -----------------------------------
__global__ void zero_f32_kernel(float* __restrict__ p, int n) {
    int i = blockIdx.x * blockDim.x + threadIdx.x;
    if (i < n) p[i] = 0.0f;
}

// ------------------------------- SpMM --------------------------------------
// out[row[e], :] += val[e] * xin[col[e], :]
// 8 lanes per edge; each lane: one float4 gather (L2-resident, x fits in the
// 192MB L2) + 4 global_atomic_add_f32 scatters (also L2-resident).
__global__ void spmm_scatter_kernel(const int* __restrict__ erow,
                                    const int* __restrict__ ecol,
                                    const float* __restrict__ eval,
                                    const float* __restrict__ xin,
                                    float* __restrict__ xout,
                                    int n_edges) {
    int gid = blockIdx.x * blockDim.x + threadIdx.x;
    int e   = gid >> 3;               // 8 threads per edge
    if (e >= n_edges) return;
    int fb  = (gid & 7) * 4;          // feature base: 0,4,...,28

    int   r = erow[e];
    int   c = ecol[e];
    float v = eval[e];

    const float4 xv = *reinterpret_cast<const float4*>(xin + (size_t)c * D_IN + fb);
    float* outp = xout + (size_t)r * D_IN + fb;
    atomicAdd(outp + 0, v * xv.x);
    atomicAdd(outp + 1, v * xv.y);
    atomicAdd(outp + 2, v * xv.z);
    atomicAdd(outp + 3, v * xv.w);
}

// ------------------------- fp32 WMMA GEMM ----------------------------------
// out[M=100000, 64] = X[M, 32] @ W[32, 64] + b
// One wave per 16x16 output tile; K-loop of 8 x V_WMMA_F32_16X16X4_F32.
// Tile grid: 6250 (M) x 4 (N) = 25000 tiles = 3125 blocks x 8 waves exactly,
// so EXEC is all-ones for every WMMA (required) and no bounds checks needed.
__global__ void __launch_bounds__(256)
gemm_wmma_f32_kernel(const float* __restrict__ X,
                     const float* __restrict__ W,
                     const float* __restrict__ bias,
                     float* __restrict__ out) {
    const int lane    = threadIdx.x & 31;
    const int wave    = threadIdx.x >> 5;
    const int tile    = blockIdx.x * 8 + wave;   // 8 waves / block
    const int mtile   = tile >> 2;               // 0..6249
    const int ntile   = tile & 3;                // 0..3
    const int halfSel = lane >> 4;               // 0: lanes 0-15, 1: lanes 16-31
    const int l16     = lane & 15;

    const int m = mtile * 16 + l16;              // A row for this lane
    const int n = ntile * 16 + l16;              // B/C/D column for this lane

    v8f c = {};
#pragma unroll
    for (int kb = 0; kb < D_IN; kb += 4) {
        // A 16x4 f32 frag: lane holds row m, K = kb + 2*halfSel + {0,1}
        const int klo = kb + halfSel * 2;
        const float2 av = *reinterpret_cast<const float2*>(X + (size_t)m * D_IN + klo);
        v2f a; a.x = av.x; a.y = av.y;
        // B 4x16 f32 frag: VGPR0 -> row klo, VGPR1 -> row klo+1, col n
        v2f b; b.x = W[klo * D_OUT + n]; b.y = W[(klo + 1) * D_OUT + n];

        c = __builtin_amdgcn_wmma_f32_16x16x4_f32(
                /*neg_a=*/false, a, /*neg_b=*/false, b,
                /*c_mod=*/(short)0, c, /*reuse_a=*/false, /*reuse_b=*/false);
    }

    const float bv = bias[n];
#pragma unroll
    for (int v = 0; v < 8; ++v) {
        const int row = mtile * 16 + v + halfSel * 8;   // C/D layout
        out[(size_t)row * D_OUT + n] = c[v] + bv;
    }
}

// ------------------------------ launcher -----------------------------------
extern "C" void kernel_launch(void* const* d_in, const int* in_sizes, int n_in,
                              void* d_out, int out_size, void* d_ws, size_t ws_size,
                              hipStream_t stream) {
    const float* x    = (const float*)d_in[0];
    const int*   erow = (const int*)  d_in[1];
    const int*   ecol = (const int*)  d_in[2];
    const float* eval = (const float*)d_in[3];
    const float* Wm   = (const float*)d_in[4];
    const float* bias = (const float*)d_in[5];
    // d_in[6] is k (static python int 4 in the reference) -> hardcoded hops.

    const int n_edges = in_sizes[1];

    float* buf0 = (float*)d_ws;                          // 100000*32 floats
    float* buf1 = buf0 + (size_t)N_NODES * D_IN;         // second ping-pong buf

    const int nfeat       = N_NODES * D_IN;              // 3.2M
    const int zeroBlocks  = (nfeat + 255) / 256;
    const int spmmThreads = n_edges * 8;
    const int spmmBlocks  = (spmmThreads + 255) / 256;

    // hop 1: x -> buf0
    zero_f32_kernel<<<zeroBlocks, 256, 0, stream>>>(buf0, nfeat);
    spmm_scatter_kernel<<<spmmBlocks, 256, 0, stream>>>(erow, ecol, eval, x, buf0, n_edges);
    // hop 2: buf0 -> buf1
    zero_f32_kernel<<<zeroBlocks, 256, 0, stream>>>(buf1, nfeat);
    spmm_scatter_kernel<<<spmmBlocks, 256, 0, stream>>>(erow, ecol, eval, buf0, buf1, n_edges);
    // hop 3: buf1 -> buf0
    zero_f32_kernel<<<zeroBlocks, 256, 0, stream>>>(buf0, nfeat);
    spmm_scatter_kernel<<<spmmBlocks, 256, 0, stream>>>(erow, ecol, eval, buf1, buf0, n_edges);
    // hop 4: buf0 -> buf1
    zero_f32_kernel<<<zeroBlocks, 256, 0, stream>>>(buf1, nfeat);
    spmm_scatter_kernel<<<spmmBlocks, 256, 0, stream>>>(erow, ecol, eval, buf0, buf1, n_edges);

    // final GEMM: buf1 @ W + b -> out   (25000 tiles / 8 waves per block)
    gemm_wmma_f32_kernel<<<3125, 256, 0, stream>>>(buf1, Wm, bias, (float*)d_out);
}